// NeighborNoiser_1005022347860
// MI455X (gfx1250) — compile-verified
//
#include <hip/hip_runtime.h>
#include <cstdint>

#define WW  1024   // image width (floats per row)
#define IMH 1024   // image height (rows per plane)

// ---------------------------------------------------------------------------
// Threefry-2x32 with key = jax.random.key(42) -> (k0,k1) = (0,42).
// Matches jax._src.prng.threefry2x32: 5 groups of 4 rounds, rotations
// [13,15,26,6] / [17,29,16,24], key injection after each group.
// ---------------------------------------------------------------------------
__device__ __forceinline__ uint32_t rotl32(uint32_t x, uint32_t d) {
  return (x << d) | (x >> (32u - d));   // -> v_alignbit_b32
}

__device__ __forceinline__ void threefry2x32_k0_42(uint32_t x0, uint32_t x1,
                                                   uint32_t& y0, uint32_t& y1) {
  const uint32_t ks0 = 0u;
  const uint32_t ks1 = 42u;
  const uint32_t ks2 = 0x1BD11BDAu ^ 0u ^ 42u;
  x0 += ks0; x1 += ks1;
#define TF_ROUND(r) { x0 += x1; x1 = rotl32(x1, (r)); x1 ^= x0; }
  TF_ROUND(13) TF_ROUND(15) TF_ROUND(26) TF_ROUND(6)
  x0 += ks1; x1 += ks2 + 1u;
  TF_ROUND(17) TF_ROUND(29) TF_ROUND(16) TF_ROUND(24)
  x0 += ks2; x1 += ks0 + 2u;
  TF_ROUND(13) TF_ROUND(15) TF_ROUND(26) TF_ROUND(6)
  x0 += ks0; x1 += ks1 + 3u;
  TF_ROUND(17) TF_ROUND(29) TF_ROUND(16) TF_ROUND(24)
  x0 += ks1; x1 += ks2 + 4u;
  TF_ROUND(13) TF_ROUND(15) TF_ROUND(26) TF_ROUND(6)
  x0 += ks2; x1 += ks0 + 5u;
#undef TF_ROUND
  y0 = x0; y1 = x1;
}

// jax uniform [0,1) for f32: bitcast((bits>>9) | 0x3f800000) - 1.0
__device__ __forceinline__ float u01(uint32_t b) {
  return __uint_as_float((b >> 9) | 0x3f800000u) - 1.0f;
}

// ---------------------------------------------------------------------------
// CDNA5 async global->LDS copy (ASYNCcnt path). VDST VGPR holds the
// wave-relative LDS byte offset (low 32 bits of the generic shared address).
// ---------------------------------------------------------------------------
__device__ __forceinline__ void async_ld_b128(uint32_t lds_off, const void* gaddr) {
  asm volatile("global_load_async_to_lds_b128 %0, %1, off"
               :: "v"(lds_off), "v"((uint64_t)(uintptr_t)gaddr)
               : "memory");
}

__device__ __forceinline__ void wait_async0() {
  asm volatile("s_wait_asynccnt 0" ::: "memory");
}

// ---------------------------------------------------------------------------
// One block per image row (B*C*H = 24576 blocks). 256 threads, 4 pixels each.
// Stage rows (y-1, y, y+1) of the current plane into LDS asynchronously,
// then blend the 4 clamped-shift neighbors with threefry uniforms.
// ---------------------------------------------------------------------------
__global__ __launch_bounds__(256) void neighbor_noiser_kernel(
    const float* __restrict__ t, float* __restrict__ out, uint32_t N) {
  __shared__ float s_up[WW];
  __shared__ float s_mid[WW];
  __shared__ float s_dn[WW];

  const uint32_t row    = blockIdx.x;        // global row index in [0, B*C*H)
  const uint32_t y      = row & (IMH - 1u);  // row within plane
  const uint32_t tid    = threadIdx.x;
  const uint32_t row_up = (y == 0u)          ? row : row - 1u;  // replicate edge
  const uint32_t row_dn = (y == (IMH - 1u))  ? row : row + 1u;

  // Async-stage 3 rows (4 KB each) into LDS: 16 bytes per thread per row.
  const uint32_t boff = tid * 16u;
  const char*    tb   = (const char*)t;
  async_ld_b128((uint32_t)(uintptr_t)s_up  + boff, tb + (size_t)row_up * (WW * 4u) + boff);
  async_ld_b128((uint32_t)(uintptr_t)s_mid + boff, tb + (size_t)row    * (WW * 4u) + boff);
  async_ld_b128((uint32_t)(uintptr_t)s_dn  + boff, tb + (size_t)row_dn * (WW * 4u) + boff);
  wait_async0();          // my wave's async writes have landed in LDS
  __syncthreads();        // ... and so have everyone else's

  const uint32_t x0 = tid * 4u;
  const uint32_t j0 = row * WW + x0;        // flat element index of first pixel
  const uint32_t N2 = N * 2u;               // 4N = 100,663,296 fits in u32
  const uint32_t N3 = N2 + N;

  // mid row with left/right replicate-clamped halo
  const float  m0  = s_mid[(x0 == 0u) ? 0u : (x0 - 1u)];
  const float4 mid = *(const float4*)(&s_mid[x0]);
  const float  m5  = s_mid[(x0 + 4u >= WW) ? (WW - 1u) : (x0 + 4u)];
  const float4 upv = *(const float4*)(&s_up[x0]);
  const float4 dnv = *(const float4*)(&s_dn[x0]);

  const float mm[6] = {m0, mid.x, mid.y, mid.z, mid.w, m5};
  const float uu[4] = {upv.x, upv.y, upv.z, upv.w};
  const float dd[4] = {dnv.x, dnv.y, dnv.z, dnv.w};

  float4 res;
  float* rp = (float*)&res;
#pragma unroll
  for (int p = 0; p < 4; ++p) {
    const uint32_t j = j0 + (uint32_t)p;
    // r is (4, N): up_r[j]=bits[j], down_r[j]=bits[N+j],
    // left_r[j]=bits[2N+j], right_r[j]=bits[3N+j]; JAX pairs counter i<2N
    // with i+2N and returns (y0, y1) for the two halves.
    uint32_t b_up, b_left, b_dn, b_right;
    threefry2x32_k0_42(j,      N2 + j, b_up, b_left);
    threefry2x32_k0_42(N + j,  N3 + j, b_dn, b_right);
    const float r_up = u01(b_up),   r_dn = u01(b_dn);
    const float r_l  = u01(b_left), r_r  = u01(b_right);
    const float num = uu[p] * r_up + dd[p] * r_dn + mm[p] * r_l + mm[p + 2] * r_r;
    const float den = (r_up + r_dn) + (r_l + r_r);
    rp[p] = num / den;   // IEEE divide to match jnp
  }
  *(float4*)(&out[j0]) = res;
}

extern "C" void kernel_launch(void* const* d_in, const int* in_sizes, int n_in,
                              void* d_out, int out_size, void* d_ws, size_t ws_size,
                              hipStream_t stream) {
  (void)n_in; (void)out_size; (void)d_ws; (void)ws_size;
  const float* t   = (const float*)d_in[0];
  float*       out = (float*)d_out;
  const uint32_t N    = (uint32_t)in_sizes[0];   // 8*3*1024*1024
  const uint32_t rows = N / WW;                  // B*C*H = 24576
  neighbor_noiser_kernel<<<dim3(rows), dim3(256), 0, stream>>>(t, out, N);
}